// TopologyEncoder_89781996355972
// MI455X (gfx1250) — compile-verified
//
#include <hip/hip_runtime.h>

typedef __attribute__((ext_vector_type(2))) float v2f;
typedef __attribute__((ext_vector_type(8))) float v8f;

#define HIDDEN 64

// ---------------------------------------------------------------------------
// x[n, :] = emb[z[n], :]   (one float4 per thread; 32-bit offsets)
// ---------------------------------------------------------------------------
__global__ void __launch_bounds__(256) embed_kernel(const int* __restrict__ z,
                                                    const float* __restrict__ emb,
                                                    float* __restrict__ x,
                                                    int n_nodes) {
  int tid = blockIdx.x * 256 + threadIdx.x;          // < n_nodes*16 (1.6M)
  if (tid >= n_nodes * 16) return;
  int n  = tid >> 4;
  int h4 = (tid & 15) * 4;
  int t  = z[n];
  const float4* s = (const float4*)(emb + (unsigned)(t * HIDDEN + h4));
  *(float4*)(x + (unsigned)(n * HIDDEN + h4)) = *s;
}

// ---------------------------------------------------------------------------
// agg[row[e], :] += x[col[e], :]   (16 threads per edge, float4 each;
// agg was pre-initialized with x, so it accumulates x + sum of messages).
// x (25.6 MB) and agg (25.6 MB) are both L2-resident on MI455X (192 MB L2),
// so the random gathers and the 76.8M fp32 atomics/layer run at L2 speed.
// ---------------------------------------------------------------------------
__global__ void __launch_bounds__(256) scatter_kernel(const int* __restrict__ rowi,
                                                      const int* __restrict__ coli,
                                                      const float* __restrict__ x,
                                                      float* __restrict__ agg,
                                                      int n_edges) {
  int tid = blockIdx.x * 256 + threadIdx.x;          // < n_edges*16 (19.2M)
  if (tid >= n_edges * 16) return;
  int e  = tid >> 4;
  int c4 = (tid & 15) * 4;
  int s  = coli[e];
  int d  = rowi[e];
  float4 v = *(const float4*)(x + (unsigned)(s * HIDDEN + c4));
  float* dp = agg + (unsigned)(d * HIDDEN + c4);
  unsafeAtomicAdd(dp + 0, v.x);   // global_atomic_add_f32
  unsafeAtomicAdd(dp + 1, v.y);
  unsafeAtomicAdd(dp + 2, v.z);
  unsafeAtomicAdd(dp + 3, v.w);
}

// ---------------------------------------------------------------------------
// xout = relu(agg @ W^T + b), W is [64(out), 64(in)] row-major.
// One wave32 per 16-node tile. Exact fp32 via V_WMMA_F32_16X16X4_F32:
//   A (16x4):  lane -> (m = lane&15, kg = lane>>4); vgpr v = A[m][4*kc+2*kg+v]
//   B (4x16):  B[k][o] = W[o][k]; lane o = lane&15; vgpr v = W[o][4*kc+2*kg+v]
//   C/D(16x16): vgpr v -> row (v + 8*kg), col = lane&15
// K = 64 -> 16 chained WMMAs per output tile, 4 output tiles (64 out feats).
// EXEC stays all-ones through the WMMAs; the tail tile takes a uniform
// (scalar-branch) slow path so the common case has branch-free stores.
// ---------------------------------------------------------------------------
__global__ void __launch_bounds__(32) gnn_gemm_wmma(const float* __restrict__ agg,
                                                    const float* __restrict__ Wl,
                                                    const float* __restrict__ bl,
                                                    float* __restrict__ xout,
                                                    int n_nodes) {
  int lane  = threadIdx.x;
  int m     = lane & 15;
  int kg    = lane >> 4;                 // 0 or 1
  int node0 = blockIdx.x * 16;
  bool full = (node0 + 16 <= n_nodes);   // uniform across the wave

  // Load A fragments for the whole K=64 strip (16 chunks of K=4).
  int an = node0 + m;
  if (an >= n_nodes) an = n_nodes - 1;   // clamp (tail only): EXEC stays full
  const float* arow = agg + (unsigned)(an * HIDDEN);
  v2f a[16];
#pragma unroll
  for (int kc = 0; kc < 16; ++kc) {
    int k = kc * 4 + kg * 2;
    a[kc].x = arow[k];
    a[kc].y = arow[k + 1];
  }

#pragma unroll
  for (int ot = 0; ot < 4; ++ot) {
    float bias = bl[ot * 16 + m];
    v8f c;
#pragma unroll
    for (int v = 0; v < 8; ++v) c[v] = bias;   // bias depends on col only

    const float* wrow = Wl + (unsigned)((ot * 16 + m) * HIDDEN);
#pragma unroll
    for (int kc = 0; kc < 16; ++kc) {
      int k = kc * 4 + kg * 2;
      v2f bf;
      bf.x = wrow[k];
      bf.y = wrow[k + 1];
      // 8 args: (neg_a, A, neg_b, B, c_mod, C, reuse_a, reuse_b)
      c = __builtin_amdgcn_wmma_f32_16x16x4_f32(false, a[kc], false, bf,
                                                (short)0, c, false, false);
    }

    int col = ot * 16 + m;
    if (full) {
      // Branch-free stores: 8 coalesced b32 stores per lane.
#pragma unroll
      for (int v = 0; v < 8; ++v) {
        float r = c[v] > 0.0f ? c[v] : 0.0f;
        xout[(unsigned)((node0 + v + 8 * kg) * HIDDEN + col)] = r;
      }
    } else {
#pragma unroll
      for (int v = 0; v < 8; ++v) {
        int nrow = node0 + v + 8 * kg;
        if (nrow < n_nodes) {
          float r = c[v] > 0.0f ? c[v] : 0.0f;
          xout[(unsigned)(nrow * HIDDEN + col)] = r;
        }
      }
    }
  }
}

// ---------------------------------------------------------------------------
extern "C" void kernel_launch(void* const* d_in, const int* in_sizes, int n_in,
                              void* d_out, int out_size, void* d_ws, size_t ws_size,
                              hipStream_t stream) {
  const int*   z    = (const int*)d_in[0];
  const int*   edge = (const int*)d_in[1];   // [2, E]: row = edge[0:E], col = edge[E:2E]
  const float* emb  = (const float*)d_in[2];
  const float* W    = (const float*)d_in[3]; // [L, 64, 64]
  const float* b    = (const float*)d_in[4]; // [L, 64]
  float*       out  = (float*)d_out;

  int n_nodes  = in_sizes[0];
  int n_edges  = in_sizes[1] / 2;
  int n_layers = in_sizes[3] / (HIDDEN * HIDDEN);

  float* x   = (float*)d_ws;
  float* agg = x + (size_t)n_nodes * HIDDEN;
  size_t xbytes = (size_t)n_nodes * HIDDEN * sizeof(float);

  {
    int blocks = (n_nodes * 16 + 255) / 256;
    embed_kernel<<<blocks, 256, 0, stream>>>(z, emb, x, n_nodes);
  }

  int ntiles = (n_nodes + 15) / 16;
  for (int i = 0; i < n_layers; ++i) {
    // agg <- x (residual folded in), then scatter-add messages into agg.
    hipMemcpyAsync(agg, x, xbytes, hipMemcpyDeviceToDevice, stream);
    int blocks = (n_edges * 16 + 255) / 256;
    scatter_kernel<<<blocks, 256, 0, stream>>>(edge, edge + n_edges, x, agg, n_edges);

    float* xout = (i == n_layers - 1) ? out : x;
    gnn_gemm_wmma<<<ntiles, 32, 0, stream>>>(agg,
                                             W + (size_t)i * HIDDEN * HIDDEN,
                                             b + (size_t)i * HIDDEN,
                                             xout, n_nodes);
  }
}